// SVGPG_Layer_4320737100003
// MI455X (gfx1250) — compile-verified
//
#include <hip/hip_runtime.h>
#include <hip/hip_bf16.h>
#include <stdint.h>

// Problem constants (from reference): N=2048, NODES=64, DIN=4, DOUT=4, M=128, P=32
#define NTOT   2048
#define NODES  64
#define DOUTC  4
#define MM     128
#define PP     32
#define LDP    132      // padded LDS row stride for prep-kernel matrices
#define LDM    130      // TDM-padded LDS row stride for Minv/Lq tiles (floats)
#define LDK2   260      // pair-interleaved stride: row-pair r>>1, elem 2n+(r&1)
#define JITTERV 1.0e-4f

typedef __attribute__((ext_vector_type(2))) float v2f;
typedef __attribute__((ext_vector_type(8))) float v8f;
typedef __attribute__((ext_vector_type(4))) unsigned int v4u;
typedef __attribute__((ext_vector_type(4))) int v4i;
typedef __attribute__((ext_vector_type(8))) int v8i;

__device__ __forceinline__ v8f wmma_f32_16x16x4(v2f a, v2f b, v8f c) {
  return __builtin_amdgcn_wmma_f32_16x16x4_f32(false, a, false, b, (short)0, c,
                                               false, false);
}

__device__ __forceinline__ v8f zero8() {
  v8f z;
#pragma unroll
  for (int i = 0; i < 8; ++i) z[i] = 0.0f;
  return z;
}

// Pair-interleaved LDS matrix: element (r, c) at (r>>1)*LDK2 + 2c + (r&1).
// A B-fragment (rows r,r+1 with r even, col c) is one aligned 8-byte load.
__device__ __forceinline__ v2f ldsPairLoad(const float* base, int rEven, int c) {
  return *(const v2f*)(base + (rEven >> 1) * LDK2 + 2 * c);
}
__device__ __forceinline__ void ldsPairStore(float* base, int rEven, int c,
                                             float lo, float hi) {
  v2f p; p.x = lo; p.y = hi;
  *(v2f*)(base + (rEven >> 1) * LDK2 + 2 * c) = p;
}

// ---------------------------------------------------------------------------
// TDM: async-load one contiguous 128x128 f32 tile (16384 dwords) from global
// into LDS, 2 pad dwords after every 128 dwords -> LDS row stride 130 (even,
// non-multiple-of-64: aligned v2f A-frag reads, no bank collisions).
// D# per CDNA5 ISA ch.8. Tracked by TENSORcnt; issue from one wave.
// ---------------------------------------------------------------------------
__device__ __forceinline__ void tdm_load_tile_padded(const float* gsrc,
                                                     const float* lds_dst) {
  const uint64_t ga = (uint64_t)(uintptr_t)gsrc;
  const uint32_t lo = (uint32_t)(uintptr_t)lds_dst;  // LDS byte offset
  v4u g0;
  g0[0] = 1u;                                            // count = 1
  g0[1] = lo;                                            // lds_addr (bytes)
  g0[2] = (uint32_t)ga;                                  // global_addr lo
  g0[3] = ((uint32_t)(ga >> 32) & 0x01FFFFFFu) | 0x80000000u;  // hi | type=2
  v8i g1;
  g1[0] = (int)0x03920000u;  // pad_amount=1(2dw)<<25 | pad_int=6(128dw)<<22
                             // | pad_en<<20 | data_size=2(4B)<<16 | wgmask=0
  g1[1] = (int)0x40000000u;  // tensor_dim0 = 16384 (low16 of bits 79:48)
  g1[2] = (int)0x00010000u;  // tensor_dim1 = 1
  g1[3] = (int)0x40000000u;  // tile_dim0 = 16384
  g1[4] = 1;                 // tile_dim1 = 1, tile_dim2 = 0
  g1[5] = 16384;             // tensor_dim0_stride (data_size units)
  g1[6] = 0;
  g1[7] = 0;
  v4i z4 = {0, 0, 0, 0};
#if defined(__clang_major__) && (__clang_major__ >= 23)
  v8i z8 = {0, 0, 0, 0, 0, 0, 0, 0};
  __builtin_amdgcn_tensor_load_to_lds(g0, g1, z4, z4, z8, 0);
#else
  __builtin_amdgcn_tensor_load_to_lds(g0, g1, z4, z4, 0);
#endif
}

// ---------------------------------------------------------------------------
// Kernel 1: per node -> gather Zn, Ku = rbf+jitter, Cholesky, L^-1,
//           Minv = L^-T L^-1 -> scratch. 128 threads / node.
// ---------------------------------------------------------------------------
__global__ __launch_bounds__(128)
void svgp_prep_kernel(const float* __restrict__ Z, const int* __restrict__ pa,
                      float* __restrict__ MinvW, float* __restrict__ ZnW,
                      float* __restrict__ znW) {
  extern __shared__ float smem[];
  float* Zn  = smem;                 // 128*32
  float* zn2 = Zn + MM * PP;         // 128
  float* Ku  = zn2 + MM;             // 128*LDP
  float* Li  = Ku + MM * LDP;        // 128*LDP
  int*   paL = (int*)(Li + MM * LDP);

  const int t = threadIdx.x;
  const int g = blockIdx.x;

  if (t < PP) paL[t] = pa[g * PP + t];
  __syncthreads();

  {
    float s = 0.0f;
#pragma unroll
    for (int p = 0; p < PP; ++p) {
      float v = Z[(size_t)t * (NODES * 4) + paL[p]];
      Zn[t * PP + p] = v;
      s += v * v;
    }
    zn2[t] = s;
    znW[(size_t)g * MM + t] = s;
  }
  __syncthreads();
  for (int i = t; i < MM * PP; i += 128) ZnW[(size_t)g * MM * PP + i] = Zn[i];

  {
    float zr[PP];
#pragma unroll
    for (int p = 0; p < PP; ++p) zr[p] = Zn[t * PP + p];
    const float z2 = zn2[t];
    for (int j = 0; j < MM; ++j) {
      float dot = 0.0f;
#pragma unroll
      for (int p = 0; p < PP; ++p) dot = fmaf(zr[p], Zn[j * PP + p], dot);
      float sq = fmaxf(z2 + zn2[j] - 2.0f * dot, 0.0f);
      float v = expf(-0.5f * sq);
      if (j == t) v += JITTERV;
      Ku[t * LDP + j] = v;
    }
  }
  __syncthreads();

  for (int k = 0; k < MM; ++k) {
    if (t == k) Ku[k * LDP + k] = sqrtf(Ku[k * LDP + k]);
    __syncthreads();
    if (t > k) Ku[t * LDP + k] /= Ku[k * LDP + k];
    __syncthreads();
    if (t > k) {
      const float lik = Ku[t * LDP + k];
      for (int j = k + 1; j <= t; ++j)
        Ku[t * LDP + j] = fmaf(-lik, Ku[j * LDP + k], Ku[t * LDP + j]);
    }
    __syncthreads();
  }

  for (int r = 0; r < MM; ++r) {          // Li = L^-1, thread t = column t
    float v = 0.0f;
    if (r >= t) {
      float s = (r == t) ? 1.0f : 0.0f;
      for (int j = t; j < r; ++j) s = fmaf(-Ku[r * LDP + j], Li[j * LDP + t], s);
      v = s / Ku[r * LDP + r];
    }
    Li[r * LDP + t] = v;
  }
  __syncthreads();

  for (int j = 0; j < MM; ++j) {          // Minv = Li^T Li, thread t = row t
    const int k0 = (t > j) ? t : j;
    float s = 0.0f;
    for (int k = k0; k < MM; ++k) s = fmaf(Li[k * LDP + t], Li[k * LDP + j], s);
    MinvW[(size_t)g * MM * MM + t * MM + j] = s;
  }
}

// ---------------------------------------------------------------------------
// Kernel 2: per (node, 128-row X tile). TDM streams Minv/Lq_d tiles into LDS
// (double-buffered), WMMA f32 does the K=128 GEMMs. Kuf/A stored
// pair-interleaved so every B-fragment is a single aligned ds_load_b64.
// Block = 256 threads = 8 wave32; grid = (16, 64).
// ---------------------------------------------------------------------------
__global__ __launch_bounds__(256)
void svgp_main_kernel(const float* __restrict__ X, const float* __restrict__ q_mu,
                      const float* __restrict__ q_sqrt, const int* __restrict__ pa,
                      const float* __restrict__ MinvW, const float* __restrict__ ZnW,
                      const float* __restrict__ znW, float* __restrict__ out) {
  extern __shared__ float smem[];
  float* bufA  = smem;                        // 128*LDM (TDM: Minv, then Lq)
  float* bufB  = bufA + MM * LDM;             // 128*LDM (TDM: Lq ping-pong)
  float* KufP  = bufB + MM * LDM;             // (MM/2)*LDK2 pair-interleaved
  float* A_sP  = KufP + (MM / 2) * LDK2;      // (MM/2)*LDK2 pair-interleaved
  float* Zn    = A_sP + (MM / 2) * LDK2;      // 128*32
  float* Xn    = Zn + MM * PP;                // 128*32
  float* zn2   = Xn + MM * PP;                // 128
  float* xn2   = zn2 + MM;                    // 128
  float* qmu_s = xn2 + MM;                    // 128*4
  float* colAK = qmu_s + MM * DOUTC;          // 128
  float* delta = colAK + MM;                  // 4*128
  int*   paL   = (int*)(delta + DOUTC * MM);

  const int t    = threadIdx.x;
  const int node = blockIdx.y;
  const int n0   = blockIdx.x * MM;
  const int lane = t & 31;
  const int w    = t >> 5;
  const int l15  = lane & 15;
  const int kk   = (lane >> 4) << 1;          // f32 WMMA frag K split: 0 or 2

  // Kick off async TDM load of Minv tile into bufA; overlaps stages 1-2.
  if (w == 0)
    tdm_load_tile_padded(MinvW + (size_t)node * MM * MM, bufA);

  if (t < PP) paL[t] = pa[node * PP + t];
  for (int i = t; i < DOUTC * MM; i += 256) delta[i] = 0.0f;
  __syncthreads();

  // Stage 1: Zn/zn2 from scratch, gather Xn tile, qmu
  for (int i = t; i < MM * PP; i += 256) Zn[i] = ZnW[(size_t)node * MM * PP + i];
  if (t < MM) zn2[t] = znW[(size_t)node * MM + t];
  for (int i = t; i < MM * PP; i += 256) {
    const int n = i >> 5, p = i & 31;
    Xn[i] = X[(size_t)(n0 + n) * (NODES * 4) + paL[p]];
  }
  for (int i = t; i < MM * DOUTC; i += 256)
    qmu_s[i] = q_mu[(size_t)(i >> 2) * (NODES * DOUTC) + node * DOUTC + (i & 3)];
  __syncthreads();
  if (t < MM) {
    float s = 0.0f;
#pragma unroll
    for (int p = 0; p < PP; ++p) s = fmaf(Xn[t * PP + p], Xn[t * PP + p], s);
    xn2[t] = s;
  }
  __syncthreads();

  // Stage 2: Kuf[m][n] = exp(-0.5*max(|z|^2+|x|^2-2 z.x, 0)), pair-interleaved
  {
    const int m = t & 127;
    const int nbeg = (t >> 7) * 64;
    float zr[PP];
#pragma unroll
    for (int p = 0; p < PP; ++p) zr[p] = Zn[m * PP + p];
    const float z2 = zn2[m];
    float* krow = KufP + (m >> 1) * LDK2 + (m & 1);
    for (int n = nbeg; n < nbeg + 64; ++n) {
      float dot = 0.0f;
#pragma unroll
      for (int p = 0; p < PP; ++p) dot = fmaf(zr[p], Xn[n * PP + p], dot);
      const float sq = fmaxf(z2 + xn2[n] - 2.0f * dot, 0.0f);
      krow[2 * n] = expf(-0.5f * sq);
    }
  }
  __syncthreads();
  if (w == 0) __builtin_amdgcn_s_wait_tensorcnt(0);  // Minv tile landed
  __syncthreads();

  // Stage 3: A = Minv @ Kuf via V_WMMA_F32_16X16X4_F32 (Minv from LDS bufA)
  {
    v8f acc[8];
#pragma unroll
    for (int tc = 0; tc < 8; ++tc) acc[tc] = zero8();
    const int rbase = (w * 16 + l15) * LDM;   // even stride -> 8B-aligned v2f
    for (int k = 0; k < MM; k += 4) {
      const v2f a = *(const v2f*)(bufA + rbase + k + kk);
#pragma unroll
      for (int tc = 0; tc < 8; ++tc) {
        const v2f b = ldsPairLoad(KufP, k + kk, tc * 16 + l15);
        acc[tc] = wmma_f32_16x16x4(a, b, acc[tc]);
      }
    }
#pragma unroll
    for (int tc = 0; tc < 8; ++tc)
#pragma unroll
      for (int v = 0; v < 8; v += 2) {  // rows (v,v+1): adjacent VGPRs+addrs
        const int row = w * 16 + ((lane >> 4) << 3) + v;
        ldsPairStore(A_sP, row, tc * 16 + l15, acc[tc][v], acc[tc][v + 1]);
      }
  }
  __syncthreads();

  // Prefetch Lq_0 tile into bufB while stage 4 runs on VALU.
  const float* qsBase = q_sqrt + (size_t)node * DOUTC * MM * MM;
  if (w == 0) tdm_load_tile_padded(qsBase, bufB);

  // Stage 4: colAK[n] = sum_m A*Kuf (== a^T Ku a); mean = A^T qmu
  if (t < MM) {
    const int n = t;
    float cak = 0.0f, m0 = 0.0f, m1 = 0.0f, m2 = 0.0f, m3 = 0.0f;
    for (int pr = 0; pr < MM / 2; ++pr) {     // row pair (2pr, 2pr+1)
      const v2f a2 = *(const v2f*)(A_sP + pr * LDK2 + 2 * n);
      const v2f k2 = *(const v2f*)(KufP + pr * LDK2 + 2 * n);
      cak = fmaf(a2.x, k2.x, cak);
      cak = fmaf(a2.y, k2.y, cak);
      const float* q0 = qmu_s + (2 * pr) * 4;
      m0 = fmaf(a2.x, q0[0], fmaf(a2.y, q0[4], m0));
      m1 = fmaf(a2.x, q0[1], fmaf(a2.y, q0[5], m1));
      m2 = fmaf(a2.x, q0[2], fmaf(a2.y, q0[6], m2));
      m3 = fmaf(a2.x, q0[3], fmaf(a2.y, q0[7], m3));
    }
    colAK[n] = cak;
    float* mrow = out + (size_t)(n0 + n) * (NODES * DOUTC) + node * DOUTC;
    mrow[0] = m0; mrow[1] = m1; mrow[2] = m2; mrow[3] = m3;
  }
  __syncthreads();

  // Stage 5: per d: C = tril(Lq_d)^T @ A (WMMA, Lq from TDM'd LDS tile,
  // double-buffered), delta_d[n] += colsum(C^2) via ds_add_f32.
  for (int d = 0; d < DOUTC; ++d) {
    if (w == 0) __builtin_amdgcn_s_wait_tensorcnt(0);  // tile d ready
    __syncthreads();                                   // all waves: d ready,
                                                       // d-1 fully consumed
    if (w == 0 && d < DOUTC - 1)                       // stream tile d+1
      tdm_load_tile_padded(qsBase + (size_t)(d + 1) * MM * MM,
                           ((d + 1) & 1) ? bufA : bufB);
    const float* Lq = (d & 1) ? bufA : bufB;           // [m][k], stride LDM
    v8f acc[8];
#pragma unroll
    for (int tc = 0; tc < 8; ++tc) acc[tc] = zero8();
    const int rowK = w * 16 + l15;                     // output row k of C
    for (int k = 0; k < MM; k += 4) {
      const int m0 = k + kk;                           // inner index m
      v2f a;                                           // LqT[k][m], m >= k
      a.x = (rowK <= m0)     ? Lq[m0 * LDM + rowK]       : 0.0f;
      a.y = (rowK <= m0 + 1) ? Lq[(m0 + 1) * LDM + rowK] : 0.0f;
#pragma unroll
      for (int tc = 0; tc < 8; ++tc) {
        const v2f b = ldsPairLoad(A_sP, k + kk, tc * 16 + l15);
        acc[tc] = wmma_f32_16x16x4(a, b, acc[tc]);
      }
    }
#pragma unroll
    for (int tc = 0; tc < 8; ++tc) {
      float s = 0.0f;
#pragma unroll
      for (int v = 0; v < 8; ++v) s = fmaf(acc[tc][v], acc[tc][v], s);
      atomicAdd(&delta[d * MM + tc * 16 + l15], s);    // ds_add_f32
    }
  }
  __syncthreads();

  // Stage 6: var = 1 + ||Lq^T a||^2 - a^T Ku a
  if (t < MM) {
    const int n = t;
    float* vrow = out + (size_t)NTOT * (NODES * DOUTC) +
                  (size_t)(n0 + n) * (NODES * DOUTC) + node * DOUTC;
    const float cak = colAK[n];
#pragma unroll
    for (int d = 0; d < DOUTC; ++d)
      vrow[d] = 1.0f + delta[d * MM + n] - cak;
  }
}

// ---------------------------------------------------------------------------
extern "C" void kernel_launch(void* const* d_in, const int* in_sizes, int n_in,
                              void* d_out, int out_size, void* d_ws, size_t ws_size,
                              hipStream_t stream) {
  const float* X      = (const float*)d_in[0];   // (2048, 256)
  const float* Z      = (const float*)d_in[1];   // (128, 256)
  const float* q_mu   = (const float*)d_in[2];   // (128, 256)
  const float* q_sqrt = (const float*)d_in[3];   // (64, 4, 128, 128)
  const int*   pa     = (const int*)d_in[4];     // (64, 32)
  float* out = (float*)d_out;                    // mean ++ var

  // Scratch (floats): Minv 64*16384 | Zn 64*4096 | znorm 64*128 (~5.3MB)
  float* MinvW = (float*)d_ws;
  float* ZnW   = MinvW + (size_t)NODES * MM * MM;
  float* znW   = ZnW + (size_t)NODES * MM * PP;

  const size_t smem1 = (size_t)(MM * PP + MM + 2 * MM * LDP) * 4 + PP * 4;
  const size_t smem2 = (size_t)(2 * MM * LDM + 2 * (MM / 2) * LDK2 +
                                2 * MM * PP + 2 * MM + MM * DOUTC + MM +
                                DOUTC * MM) * 4 +
                       PP * 4;

  svgp_prep_kernel<<<dim3(NODES), dim3(128), smem1, stream>>>(Z, pa, MinvW, ZnW, znW);
  svgp_main_kernel<<<dim3(NTOT / MM, NODES), dim3(256), smem2, stream>>>(
      X, q_mu, q_sqrt, pa, MinvW, ZnW, znW, out);
}